// HeterogeneousLightGCN_5927054868559
// MI455X (gfx1250) — compile-verified
//
#include <hip/hip_runtime.h>

#define DIM 64

typedef __attribute__((ext_vector_type(2))) float v2f;
typedef __attribute__((ext_vector_type(8))) float v8f;

// ---------------------------------------------------------------------------
// init: cur[i] = acc[i] = concat(user_emb, product_emb)[i]; deg = 0
// ---------------------------------------------------------------------------
__global__ void lgcn_init(const float* __restrict__ ue, const float* __restrict__ pe,
                          float* __restrict__ cur, float* __restrict__ acc,
                          unsigned* __restrict__ deg, int nu64, int total, int n_nodes) {
    int i = blockIdx.x * blockDim.x + threadIdx.x;
    if (i < total) {
        float v = (i < nu64) ? ue[i] : pe[i - nu64];
        cur[i] = v;
        acc[i] = v;
    }
    if (i < n_nodes) deg[i] = 0u;
}

// ---------------------------------------------------------------------------
// degree of destination nodes (col)
// ---------------------------------------------------------------------------
__global__ void lgcn_degree(const int* __restrict__ col, unsigned* __restrict__ deg, int E) {
    int e = blockIdx.x * blockDim.x + threadIdx.x;
    if (e < E) {
        int c = __builtin_nontemporal_load(&col[e]);
        atomicAdd(&deg[c], 1u);
    }
}

// ---------------------------------------------------------------------------
// dis[i] = deg>0 ? deg^-0.5 : 0
// ---------------------------------------------------------------------------
__global__ void lgcn_rsqrt(const unsigned* __restrict__ deg, float* __restrict__ dis, int n) {
    int i = blockIdx.x * blockDim.x + threadIdx.x;
    if (i < n) {
        unsigned d = deg[i];
        dis[i] = (d > 0u) ? rsqrtf((float)d) : 0.0f;
    }
}

// ---------------------------------------------------------------------------
// zero a buffer (float4 granularity)
// ---------------------------------------------------------------------------
__global__ void lgcn_zero4(float4* __restrict__ p, int n4) {
    int i = blockIdx.x * blockDim.x + threadIdx.x;
    if (i < n4) p[i] = make_float4(0.f, 0.f, 0.f, 0.f);
}

// ---------------------------------------------------------------------------
// propagate: next[col] += dis[row]*dis[col] * cur[row]
// one half-wave (16 lanes) per edge; each lane owns one float4 (4 dims)
// edge stream non-temporal (keep L2 for the embedding tables)
// scatter via hardware f32 atomics (L2-resident destinations)
// ---------------------------------------------------------------------------
__global__ void lgcn_prop(const int* __restrict__ row, const int* __restrict__ col,
                          const float* __restrict__ dis,
                          const float4* __restrict__ cur4, float* __restrict__ next,
                          int work /* = 16*E */) {
    int idx = blockIdx.x * blockDim.x + threadIdx.x;
    if (idx >= work) return;
    int e = idx >> 4;
    int l = idx & 15;
    int r = __builtin_nontemporal_load(&row[e]);
    int c = __builtin_nontemporal_load(&col[e]);
    float w = dis[r] * dis[c];
    float4 v = cur4[r * 16 + l];
    float* dst = next + (size_t)c * DIM + l * 4;
    unsafeAtomicAdd(dst + 0, w * v.x);
    unsafeAtomicAdd(dst + 1, w * v.y);
    unsafeAtomicAdd(dst + 2, w * v.z);
    unsafeAtomicAdd(dst + 3, w * v.w);
}

// ---------------------------------------------------------------------------
// acc += next via V_WMMA_F32_16X16X4_F32.
// One wave per 16x16 f32 tile. Chain 4 WMMAs with identity-selector A:
//   A_j[m,k] = (m == 4j+k),  B_j[k,n] = next_tile[4j+k, n]
//   => D = sum_j A_j*B_j + C = next_tile + acc_tile
// Layouts per CDNA5 ISA:
//   A 16x4 : VGPR0 = K0 (lanes 0-15) / K2 (lanes 16-31); VGPR1 = K1 / K3
//   B 4x16 : same split, N = lane%16
//   C/D    : VGPR v = row v (lanes 0-15) / row v+8 (lanes 16-31), N = lane%16
// Requires EXEC all-1s: grid sized exactly, no divergence (N % 16 == 0 path).
// ---------------------------------------------------------------------------
__global__ void lgcn_acc_wmma(const float* __restrict__ next, float* __restrict__ acc) {
    int wave = (blockIdx.x * blockDim.x + threadIdx.x) >> 5;
    int lane = threadIdx.x & 31;
    int tr = wave >> 2;        // row tile (16 nodes)
    int tc = wave & 3;         // col tile (16 dims of 64)
    const float* nb = next + (size_t)tr * 16 * DIM + tc * 16;
    float*       ab = acc  + (size_t)tr * 16 * DIM + tc * 16;
    int  l  = lane & 15;
    bool hi = lane >= 16;

    v8f c;
#pragma unroll
    for (int v = 0; v < 8; ++v) c[v] = ab[(v + (hi ? 8 : 0)) * DIM + l];

#pragma unroll
    for (int j = 0; j < 4; ++j) {
        int k0 = 4 * j + (hi ? 2 : 0);
        int k1 = 4 * j + (hi ? 3 : 1);
        v2f a, b;
        a.x = (l == k0) ? 1.0f : 0.0f;
        a.y = (l == k1) ? 1.0f : 0.0f;
        b.x = nb[k0 * DIM + l];
        b.y = nb[k1 * DIM + l];
        c = __builtin_amdgcn_wmma_f32_16x16x4_f32(false, a, false, b, (short)0, c,
                                                  false, false);
    }

#pragma unroll
    for (int v = 0; v < 8; ++v) ab[(v + (hi ? 8 : 0)) * DIM + l] = c[v];
}

// fallback scalar accumulate for a row remainder (not hit when N%16==0)
__global__ void lgcn_acc_tail(const float* __restrict__ next, float* __restrict__ acc,
                              int base, int total) {
    int i = blockIdx.x * blockDim.x + threadIdx.x + base;
    if (i < total) acc[i] += next[i];
}

// ---------------------------------------------------------------------------
// final: acc *= 1/(L+1)
// ---------------------------------------------------------------------------
__global__ void lgcn_scale(float4* __restrict__ p, float s, int n4) {
    int i = blockIdx.x * blockDim.x + threadIdx.x;
    if (i < n4) {
        float4 v = p[i];
        p[i] = make_float4(v.x * s, v.y * s, v.z * s, v.w * s);
    }
}

extern "C" void kernel_launch(void* const* d_in, const int* in_sizes, int n_in,
                              void* d_out, int out_size, void* d_ws, size_t ws_size,
                              hipStream_t stream) {
    const float* ue = (const float*)d_in[0];
    const float* pe = (const float*)d_in[1];
    const int*   ei = (const int*)d_in[2];

    const int NU = in_sizes[0] / DIM;
    const int NP = in_sizes[1] / DIM;
    const int N  = NU + NP;
    const int E  = in_sizes[2] / 2;
    const int* row = ei;          // edge_index[0]
    const int* col = ei + E;      // edge_index[1]

    const int total = N * DIM;    // 19.2M floats
    float* acc = (float*)d_out;

    // workspace layout: cur | next | deg | dis
    char* ws = (char*)d_ws;
    float*    bufA = (float*)ws;
    float*    bufB = (float*)(ws + (size_t)total * sizeof(float));
    unsigned* deg  = (unsigned*)(ws + 2 * (size_t)total * sizeof(float));
    float*    dis  = (float*)(ws + 2 * (size_t)total * sizeof(float) + (size_t)N * sizeof(unsigned));

    float* cur = bufA;
    float* nxt = bufB;

    const int B = 256;

    // init cur/acc/deg
    lgcn_init<<<(total + B - 1) / B, B, 0, stream>>>(ue, pe, cur, acc, deg,
                                                     NU * DIM, total, N);
    // degrees
    lgcn_degree<<<(E + B - 1) / B, B, 0, stream>>>(col, deg, E);
    // deg^-1/2
    lgcn_rsqrt<<<(N + B - 1) / B, B, 0, stream>>>(deg, dis, N);

    const int work  = E * 16;          // 80M half-wave work items
    const int n4    = total / 4;
    const int tiles = (N / 16) * 4;    // 16x16 tiles (N=300000 -> exact)
    const int rem   = (N % 16) * DIM;  // leftover rows (0 here)

    for (int layer = 0; layer < 3; ++layer) {
        lgcn_zero4<<<(n4 + B - 1) / B, B, 0, stream>>>((float4*)nxt, n4);
        lgcn_prop<<<(work + B - 1) / B, B, 0, stream>>>(row, col, dis,
                                                        (const float4*)cur, nxt, work);
        if (tiles > 0)
            lgcn_acc_wmma<<<(tiles * 32) / B, B, 0, stream>>>(nxt, acc);
        if (rem > 0)
            lgcn_acc_tail<<<(rem + B - 1) / B, B, 0, stream>>>(nxt, acc,
                                                               (N / 16) * 16 * DIM, total);
        // ping-pong
        float* t = cur; cur = nxt; nxt = t;
    }

    lgcn_scale<<<(n4 + B - 1) / B, B, 0, stream>>>((float4*)acc, 0.25f, n4);
}